// AtomPooling_7911329759781
// MI455X (gfx1250) — compile-verified
//
#include <hip/hip_runtime.h>
#include <math.h>

typedef float    v2f __attribute__((ext_vector_type(2)));
typedef float    v8f __attribute__((ext_vector_type(8)));
typedef unsigned v4u __attribute__((ext_vector_type(4)));
typedef int      v4i __attribute__((ext_vector_type(4)));
typedef int      v8i __attribute__((ext_vector_type(8)));

#define DIM        256
#define MAX_ROWS   200
#define ROWS_PAD   208
#define SEG_FLOATS (MAX_ROWS * DIM)

extern __shared__ float smem[];

__global__ __launch_bounds__(256, 1)
void atom_pool_kernel(const float* __restrict__ atom,
                      const int*   __restrict__ idx,
                      const float* __restrict__ w,
                      const float* __restrict__ bias,
                      float* __restrict__ out)
{
    const int seg  = blockIdx.x;
    const int tid  = threadIdx.x;
    const int wave = tid >> 5;
    const int lane = tid & 31;

    float* tile   = smem;                  // [200][256] segment features
    float* scores = smem + SEG_FLOATS;     // [208]
    float* att    = scores + ROWS_PAD;     // [208] exp(score - max), 0-padded
    float* red    = att + ROWS_PAD;        // [2]  {max, denom}

    const int start = idx[2 * seg];
    const int end   = idx[2 * seg + 1];
    int len = end - start + 1;
    if (len > MAX_ROWS) len = MAX_ROWS;
    const int len4 = (len + 3) & ~3;       // tile rows padded to multiple of 4
    const int qmax = len4 >> 2;

    // ---- Tensor Data Mover: DMA the len x 256 f32 tile into LDS (single HBM read) ----
    if (wave == 0) {
        unsigned long long ga =
            (unsigned long long)(const void*)(atom + (size_t)start * DIM);
        unsigned lds_off = (unsigned)(unsigned long long)(void*)tile;

        v4u g0;
        g0.x = 1u;                                       // count = 1 valid descriptor
        g0.y = lds_off;                                  // lds_addr (bytes)
        g0.z = (unsigned)(ga & 0xFFFFFFFFull);           // global_addr[31:0]
        g0.w = (unsigned)((ga >> 32) & 0x1FFFFFFull)     // global_addr[56:32]
             | (2u << 30);                               // type = 2 ("image")

        v8i g1;
        g1[0] = (int)(2u << 16);                                      // data_size = 4B
        g1[1] = (int)((unsigned)(DIM & 0xFFFF) << 16);                // tensor_dim0 lo16
        g1[2] = (int)(((unsigned)DIM >> 16)
                    | ((unsigned)(len & 0xFFFF) << 16));              // dim0 hi16 | tensor_dim1 lo16
        g1[3] = (int)((((unsigned)len >> 16) & 0xFFFFu)
                    | ((unsigned)DIM << 16));                         // dim1 hi16 | tile_dim0 = 256
        g1[4] = (int)(len & 0xFFFF);                                  // tile_dim1 = len, tile_dim2 = 0
        g1[5] = DIM;                                                  // tensor_dim0_stride lo32 = 256
        g1[6] = 0;                                                    // stride hi / dim1_stride
        g1[7] = 0;

        v4i g2 = {0, 0, 0, 0};             // 2D tensor: groups 2/3 unused
        v4i g3 = {0, 0, 0, 0};
        v8i g4 = {0, 0, 0, 0, 0, 0, 0, 0}; // extra group (6-arg toolchain form)
        __builtin_amdgcn_tensor_load_to_lds(g0, g1, g2, g3, g4, 0);
        __builtin_amdgcn_s_wait_tensorcnt(0);
    }
    __syncthreads();

    // ---- Zero-fill pad rows [len, len4) so the WMMA loop needs no bounds checks ----
    for (int i = tid; i < (len4 - len) * DIM; i += 256)
        tile[len * DIM + i] = 0.0f;

    // ---- Pass 1: scores[r] = tile[r,:] . w + b  (one wave per row, LDS-resident) ----
    const float bval = bias[0];
    const float4 w0 = ((const float4*)w)[lane];        // d = lane*4 .. +3
    const float4 w1 = ((const float4*)w)[lane + 32];   // d = 128 + lane*4 .. +3
    for (int r = wave; r < len; r += 8) {
        const float4* row = (const float4*)(tile + r * DIM);
        float4 a0 = row[lane];
        float4 a1 = row[lane + 32];
        float p = a0.x * w0.x + a0.y * w0.y + a0.z * w0.z + a0.w * w0.w
                + a1.x * w1.x + a1.y * w1.y + a1.z * w1.z + a1.w * w1.w;
        for (int m = 16; m >= 1; m >>= 1)
            p += __shfl_xor(p, m, 32);
        if (lane == 0) scores[r] = p + bval;
    }
    __syncthreads();

    // ---- Softmax: max reduce (wave 0) ----
    if (wave == 0) {
        float m = -1e30f;
        for (int r = lane; r < len; r += 32) m = fmaxf(m, scores[r]);
        for (int s = 16; s >= 1; s >>= 1) m = fmaxf(m, __shfl_xor(m, s, 32));
        if (lane == 0) red[0] = m;
    }
    __syncthreads();

    const float smax = red[0];
    for (int r = tid; r < ROWS_PAD; r += 256)
        att[r] = (r < len) ? __expf(scores[r] - smax) : 0.0f;
    __syncthreads();

    if (wave == 0) {
        float s = 0.0f;
        for (int r = lane; r < len; r += 32) s += att[r];
        for (int m = 16; m >= 1; m >>= 1) s += __shfl_xor(s, m, 32);
        if (lane == 0) red[1] = s;
    }
    __syncthreads();
    const float inv = 1.0f / red[1];

    // ---- Pass 2: pooled[d] = (1/denom) * sum_r att[r] * tile[r,d] via WMMA f32 16x16x4 ----
    // A (16x4): att[4q+k] replicated across M. Lanes 0-15 hold K={0,1}, lanes 16-31 K={2,3}.
    // B (4x16): tile[4q+k, d0+n]; accumulate over q in C. Row M=0 of D = pooled slice.
    const int half = lane >> 4;
    const int n    = lane & 15;
    for (int slice = 0; slice < 2; ++slice) {
        const int d0 = slice * 128 + wave * 16;
        const float* bp = tile + d0 + n + 2 * half * DIM;   // row 2*half, col d0+n
        v8f c = {0.f, 0.f, 0.f, 0.f, 0.f, 0.f, 0.f, 0.f};
        for (int q = 0; q < qmax; ++q) {
            const int r = q * 4 + 2 * half;
            v2f a, bm;
            a.x  = att[r];
            a.y  = att[r + 1];
            bm.x = bp[q * 4 * DIM];            // tile[r    , d0+n]
            bm.y = bp[q * 4 * DIM + DIM];      // tile[r + 1, d0+n]
            c = __builtin_amdgcn_wmma_f32_16x16x4_f32(
                    false, a, false, bm, (short)0, c, false, false);
        }
        if (half == 0)
            out[(size_t)seg * DIM + d0 + n] = c[0] * inv;
    }
}

extern "C" void kernel_launch(void* const* d_in, const int* in_sizes, int n_in,
                              void* d_out, int out_size, void* d_ws, size_t ws_size,
                              hipStream_t stream) {
    const float* atom = (const float*)d_in[0];
    const int*   idx  = (const int*)d_in[1];
    const float* w    = (const float*)d_in[2];
    const float* b    = (const float*)d_in[3];
    float* out        = (float*)d_out;

    const int n_seg = in_sizes[1] / 2;
    const size_t shmem = (size_t)(SEG_FLOATS + 2 * ROWS_PAD + 8) * sizeof(float);

    atom_pool_kernel<<<n_seg, 256, shmem, stream>>>(atom, idx, w, b, out);
    (void)d_ws; (void)ws_size; (void)n_in; (void)out_size;
}